// AFDF_41979010351586
// MI455X (gfx1250) — compile-verified
//
#include <hip/hip_runtime.h>

typedef __attribute__((ext_vector_type(2))) float v2f;
typedef __attribute__((ext_vector_type(8))) float v8f;
typedef __attribute__((ext_vector_type(4))) int   v4i;

#define FFT_N     4096
#define NTHREADS  256
#define NWAVES    8
#define INV_N     (1.0f / 4096.0f)

#if __has_builtin(__builtin_amdgcn_global_load_async_to_lds_b128)
#define HAVE_ASYNC_LDS 1
#endif

__device__ __forceinline__ v8f wmma4(v2f a, v2f b, v8f c) {
  // D(16x16) = A(16x4) * B(4x16) + C    -> v_wmma_f32_16x16x4_f32
  return __builtin_amdgcn_wmma_f32_16x16x4_f32(
      /*neg_a=*/false, a, /*neg_b=*/false, b,
      /*c_mod=*/(short)0, c, /*reuse_a=*/false, /*reuse_b=*/false);
}

// full 4096-entry table: twid[t] = exp(-2*pi*i*t/4096)
__device__ __forceinline__ float2 twload(const float2* tw, int idx) {
  return tw[idx & (FFT_N - 1)];
}

// One radix-16 stage over LDS data, batch `bidx` (16 of the 256 DFTs of this stage).
// FWD: DIF stage  (DFT with W16^-rm, then post-twiddle W_{16s}^{-j r})
// INV: DIT stage  (pre-twiddle conj W_{16s}^{j m}, then DFT with W16^{+rm})
// MULA: elementwise A-scale at gather (fwd stage 1, natural order)
// MULD: elementwise D-scale at gather with digit-reversed index (inv stage 1)
// GSTORE: scatter straight to global with 1/N (inv stage 3, natural order)
template <bool FWD, bool MULA, bool MULD, bool GSTORE>
__device__ __forceinline__ void dft_stage(float2* dbuf, const float2* twid,
                                          int sl, int bidx, int hi, int col,
                                          const v2f* fC, const v2f* fS, const v2f* fNS,
                                          const float2* __restrict__ A2,
                                          const float2* __restrict__ D2,
                                          float2* __restrict__ o2) {
  const int d     = (bidx << 4) + col;       // which DFT this lane's column belongs to
  const int s     = 1 << sl;                 // stage span: 256, 16, 1
  const int j     = d & (s - 1);
  const int g     = d >> sl;
  const int ebase = (g << (4 + sl)) + j;     // base element index for this DFT
  const int kmul  = 256 >> sl;               // 4096 / (16*s)

  // Gather the 8 rows of T this lane contributes to the B-fragments:
  // chunk kb holds rows m = 4*kb + 2*hi + {0,1} (ISA 4x16 B layout), column = col.
  float2 t[8];
#pragma unroll
  for (int kb = 0; kb < 4; ++kb) {
#pragma unroll
    for (int u = 0; u < 2; ++u) {
      const int m = 4 * kb + 2 * hi + u;
      const int e = ebase + (m << sl);
      float2 v = dbuf[e];
      if (MULA) {  // xs = A * x, real/imag scaled independently
        float2 a = A2[e];
        v.x *= a.x; v.y *= a.y;
      }
      if (MULD) {  // frequency-domain D-scale; data stored digit-reversed
        const int rv = ((e & 15) << 8) | (e & 0xF0) | (e >> 8);
        float2 dd = D2[rv];
        v.x *= dd.x; v.y *= dd.y;
      }
      if (!FWD && sl > 0) {  // DIT pre-twiddle: conj(W^{j*m*kmul}) * v
        float2 w = twload(twid, j * m * kmul);
        float re = w.x * v.x + w.y * v.y;
        float im = w.x * v.y - w.y * v.x;
        v = make_float2(re, im);
      }
      t[2 * kb + u] = v;
    }
  }

  // Complex 16x16x16 matmul via 16 chained f32 WMMAs.
  // fwd (F = C - iS): Vr = C*Tr + S*Ti ; Vi = C*Ti - S*Tr
  // inv (F = C + iS): Vr = C*Tr - S*Ti ; Vi = C*Ti + S*Tr
  v8f vr = {}, vi = {};
#pragma unroll
  for (int kb = 0; kb < 4; ++kb) {
    v2f br, bi;
    br[0] = t[2 * kb].x;     br[1] = t[2 * kb + 1].x;
    bi[0] = t[2 * kb].y;     bi[1] = t[2 * kb + 1].y;
    vr = wmma4(fC[kb], br, vr);
    vi = wmma4(fC[kb], bi, vi);
    vr = wmma4(FWD ? fS[kb] : fNS[kb], bi, vr);
    vi = wmma4(FWD ? fNS[kb] : fS[kb], br, vi);
  }

  // Scatter result rows r = v + 8*hi (ISA 16x16 D layout).
#pragma unroll
  for (int v = 0; v < 8; ++v) {
    const int r = v + 8 * hi;
    const int e = ebase + (r << sl);
    float re = vr[v], im = vi[v];
    if (GSTORE) {
      o2[e] = make_float2(re * INV_N, im * INV_N);
    } else {
      if (FWD && sl > 0) {  // DIF post-twiddle: W^{j*r*kmul}
        float2 w = twload(twid, j * r * kmul);
        float nre = re * w.x - im * w.y;
        float nim = re * w.y + im * w.x;
        re = nre; im = nim;
      }
      dbuf[e] = make_float2(re, im);
    }
  }
}

__global__ __launch_bounds__(NTHREADS) void afdf_fft_kernel(
    const float* __restrict__ xg, const float* __restrict__ Ag,
    const float* __restrict__ Dg, float* __restrict__ og) {
  __shared__ __align__(16) float2 dbuf[FFT_N];  // 32 KB: one row of complex data
  __shared__ float2 twid[FFT_N];                // 32 KB: full twiddle table

  const int tid  = threadIdx.x;
  const int lane = tid & 31;
  const int wave = tid >> 5;
  const int hi   = lane >> 4;
  const int col  = lane & 15;
  const long long row = blockIdx.x;

  const float2* x2 = (const float2*)xg + row * (long long)FFT_N;
  const float2* A2 = (const float2*)Ag;
  const float2* D2 = (const float2*)Dg;
  float2*       o2 = (float2*)og + row * (long long)FFT_N;

  // Kick off the row copy into LDS (async DMA if available), overlapped with
  // the twiddle-table transcendentals below.
#if defined(HAVE_ASYNC_LDS)
  {
    v4i* xv4 = (v4i*)x2;  // 2048 16-byte chunks
    for (int i = tid; i < FFT_N / 2; i += NTHREADS) {
      v4i* g = xv4 + i;
      v4i* l = (v4i*)(&dbuf[2 * i]);
      __builtin_amdgcn_global_load_async_to_lds_b128(g, l, 0, 0);
    }
  }
#else
  for (int i = tid; i < FFT_N; i += NTHREADS) dbuf[i] = x2[i];
#endif

  // DFT-16 matrix fragments (ISA 16x4 A layout): row = col, K = 4*kb + 2*hi + u
  v2f fC[4], fS[4], fNS[4];
#pragma unroll
  for (int kb = 0; kb < 4; ++kb) {
#pragma unroll
    for (int u = 0; u < 2; ++u) {
      const int m = 4 * kb + 2 * hi + u;
      const int p = (col * m) & 15;
      const float ang = 0.39269908169872414f * (float)p;  // 2*pi/16
      fC[kb][u]  = __builtin_cosf(ang);
      const float sv = __builtin_sinf(ang);
      fS[kb][u]  = sv;
      fNS[kb][u] = -sv;
    }
  }

  // Build full twiddle table: exp(-2*pi*i*t/4096)
  for (int t = tid; t < FFT_N; t += NTHREADS) {
    const float ang = -1.5339807878856412e-3f * (float)t;  // -2*pi/4096
    twid[t] = make_float2(__builtin_cosf(ang), __builtin_sinf(ang));
  }

#if defined(HAVE_ASYNC_LDS)
#if __has_builtin(__builtin_amdgcn_s_wait_asynccnt)
  __builtin_amdgcn_s_wait_asynccnt(0);
#else
  asm volatile("s_wait_asynccnt 0" ::: "memory");
#endif
#endif
  __syncthreads();

  // Forward FFT: DIF radix-16, spans 256/16/1 (A-scale fused into stage 1 gather).
  for (int b = wave; b < 16; b += NWAVES)
    dft_stage<true, true, false, false>(dbuf, twid, 8, b, hi, col, fC, fS, fNS, A2, D2, o2);
  __syncthreads();
  for (int b = wave; b < 16; b += NWAVES)
    dft_stage<true, false, false, false>(dbuf, twid, 4, b, hi, col, fC, fS, fNS, A2, D2, o2);
  __syncthreads();
  for (int b = wave; b < 16; b += NWAVES)
    dft_stage<true, false, false, false>(dbuf, twid, 0, b, hi, col, fC, fS, fNS, A2, D2, o2);
  __syncthreads();

  // Inverse FFT: DIT radix-16, spans 1/16/256.
  // D-scale (digit-reversed gather) fused into stage 1; 1/N + global store fused
  // into stage 3's scatter (natural order, 128B-coalesced runs).
  for (int b = wave; b < 16; b += NWAVES)
    dft_stage<false, false, true, false>(dbuf, twid, 0, b, hi, col, fC, fS, fNS, A2, D2, o2);
  __syncthreads();
  for (int b = wave; b < 16; b += NWAVES)
    dft_stage<false, false, false, false>(dbuf, twid, 4, b, hi, col, fC, fS, fNS, A2, D2, o2);
  __syncthreads();
  for (int b = wave; b < 16; b += NWAVES)
    dft_stage<false, false, false, true>(dbuf, twid, 8, b, hi, col, fC, fS, fNS, A2, D2, o2);
}

extern "C" void kernel_launch(void* const* d_in, const int* in_sizes, int n_in,
                              void* d_out, int out_size, void* d_ws, size_t ws_size,
                              hipStream_t stream) {
  (void)n_in; (void)out_size; (void)d_ws; (void)ws_size;
  const float* x = (const float*)d_in[0];
  const float* A = (const float*)d_in[1];
  const float* D = (const float*)d_in[2];
  float* out = (float*)d_out;
  const int rows = in_sizes[0] / (FFT_N * 2);  // 8192
  afdf_fft_kernel<<<dim3(rows), dim3(NTHREADS), 0, stream>>>(x, A, D, out);
}